// InvariantPointAttention_52183852646497
// MI455X (gfx1250) — compile-verified
//
#include <hip/hip_runtime.h>
#include <math.h>

typedef __attribute__((ext_vector_type(2))) float v2f;
typedef __attribute__((ext_vector_type(8))) float v8f;

#define WMMA_F32_4(A, Bv, Cc) \
  __builtin_amdgcn_wmma_f32_16x16x4_f32(false, (A), false, (Bv), (short)0, (Cc), false, false)

constexpr int Bn = 2, In = 1024, Cn = 16, Hn = 12, QPn = 4, VPn = 8;
constexpr float kWC = 0.23570226039551584f;  // sqrt(2/(9*QP)) = sqrt(1/18)
constexpr float kWL = 0.57735026918962576f;  // sqrt(1/3)
constexpr float kCSC = 0.25f;                // C^-0.5

// ---------------------------------------------------------------------------
// K1: projections + rigid transform + point norms. One block per (b,i).
// ---------------------------------------------------------------------------
__global__ __launch_bounds__(256) void k_proj(
    const float* __restrict__ s, const float* __restrict__ R, const float* __restrict__ t,
    const float* __restrict__ Wq, const float* __restrict__ Wk, const float* __restrict__ Wv,
    const float* __restrict__ Wqp, const float* __restrict__ Wkp, const float* __restrict__ Wvp,
    float* __restrict__ q, float* __restrict__ k, float* __restrict__ v,
    float* __restrict__ qpts, float* __restrict__ kpts, float* __restrict__ vpts,
    float* __restrict__ q2, float* __restrict__ k2)
{
  __shared__ float ss[16], sR[9], st[3];
  __shared__ float rq[144], rk[144], rv[288];
  const int bi = blockIdx.x;           // b*In + i
  const int tid = threadIdx.x;
  if (tid < 16)      ss[tid]      = s[bi * 16 + tid];
  else if (tid < 25) sR[tid - 16] = R[bi * 9 + (tid - 16)];
  else if (tid < 28) st[tid - 25] = t[bi * 3 + (tid - 25)];
  __syncthreads();

  for (int idx = tid; idx < 1152; idx += 256) {
    const float* W; int col, ncols; float* gdst = nullptr; float* ldst = nullptr;
    if (idx < 192)      { W = Wq;  ncols = 192; col = idx;       gdst = &q[bi * 192 + col]; }
    else if (idx < 384) { W = Wk;  ncols = 192; col = idx - 192; gdst = &k[bi * 192 + col]; }
    else if (idx < 576) { W = Wv;  ncols = 192; col = idx - 384; gdst = &v[bi * 192 + col]; }
    else if (idx < 720) { W = Wqp; ncols = 144; col = idx - 576; ldst = &rq[col]; }
    else if (idx < 864) { W = Wkp; ncols = 144; col = idx - 720; ldst = &rk[col]; }
    else                { W = Wvp; ncols = 288; col = idx - 864; ldst = &rv[col]; }
    float acc = 0.f;
#pragma unroll
    for (int r = 0; r < 16; ++r) acc += ss[r] * W[r * ncols + col];
    if (ldst) *ldst = acc; else *gdst = acc;
  }
  __syncthreads();

  if (tid < 192) {
    float* raw; int n;
    if (tid < 48)       { raw = rq; n = tid; }
    else if (tid < 96)  { raw = rk; n = tid - 48; }
    else                { raw = rv; n = tid - 96; }
    const float px = raw[n * 3 + 0], py = raw[n * 3 + 1], pz = raw[n * 3 + 2];
    const float ox = sR[0] * px + sR[1] * py + sR[2] * pz + st[0];
    const float oy = sR[3] * px + sR[4] * py + sR[5] * pz + st[1];
    const float oz = sR[6] * px + sR[7] * py + sR[8] * pz + st[2];
    if (tid < 96) {                       // q/k points: n = h*QP + p, layout (B,I,H,QP,3)
      float* g = (tid < 48) ? qpts : kpts;
      g[bi * 144 + n * 3 + 0] = ox; g[bi * 144 + n * 3 + 1] = oy; g[bi * 144 + n * 3 + 2] = oz;
      raw[n * 3 + 0] = ox; raw[n * 3 + 1] = oy; raw[n * 3 + 2] = oz;   // for q2/k2
    } else {                              // v points: n = h*VP + p, layout (B,I,H,32) padded
      const int h = n / 8, p = n % 8;
      float* g = &vpts[(size_t)bi * Hn * 32 + h * 32 + p * 3];
      g[0] = ox; g[1] = oy; g[2] = oz;
      const int u = tid - 96;             // 96 threads zero the 96 pad slots (cols 24..31)
      vpts[(size_t)bi * Hn * 32 + (u / 8) * 32 + 24 + (u % 8)] = 0.f;
    }
  }
  __syncthreads();
  if (tid < 24) {
    const int which = tid / 12, h = tid % 12;
    const float* raw = which ? rk : rq;
    float acc = 0.f;
#pragma unroll
    for (int u = 0; u < 12; ++u) { const float x = raw[h * 12 + u]; acc += x * x; }
    (which ? k2 : q2)[bi * Hn + h] = acc;
  }
}

// ---------------------------------------------------------------------------
// K2 (fused): per (b, i-tile, j-tile) block:
//   stage 1: bias = z-tile @ Wb via WMMA -> LDS (all 12 heads at once)
//   stage 2: per head h: a = W_L*(0.25*q.kT + bias + coef_h*(q2+k2-2*qp.kpT))
// `a` is written exactly once (no bias round-trip through HBM).
// ---------------------------------------------------------------------------
__global__ __launch_bounds__(256) void k_logits_fused(
    const float* __restrict__ z, const float* __restrict__ Wb,
    const float* __restrict__ q, const float* __restrict__ k,
    const float* __restrict__ qpts, const float* __restrict__ kpts,
    const float* __restrict__ q2, const float* __restrict__ k2,
    const float* __restrict__ gamma, float* __restrict__ a)
{
  __shared__ float sbias[16 * 16 * 16];   // [i_off][j_off][h]
  const int w = threadIdx.x >> 5;
  const int lane = threadIdx.x & 31;
  const int half = lane >> 4, l16 = lane & 15;
  const int blk = blockIdx.x;
  const int jt = blk & 63;
  const int it = (blk >> 6) & 63;
  const int b = blk >> 12;
  const int i0 = it * 16, j0 = jt * 16;

  const int hcl = (l16 < 12) ? l16 : 11;        // clamp + mask (branch-free loop)
  const float msk = (l16 < 12) ? 1.f : 0.f;

  // ---- stage 1: wave w handles row-tiles rt = w, w+8 (rt = i offset) ----
  for (int rt = w; rt < 16; rt += 8) {
    const float* zrow = z + (size_t)((b * In + i0 + rt) * In + j0 + l16) * 16;
    v8f c = {};
#pragma unroll
    for (int kb = 0; kb < 16; kb += 4) {
      const int kk = kb + 2 * half;
      v2f av = *reinterpret_cast<const v2f*>(zrow + kk);
      v2f bv;
      bv.x = Wb[kk * Hn + hcl] * msk;
      bv.y = Wb[(kk + 1) * Hn + hcl] * msk;
      c = WMMA_F32_4(av, bv, c);
    }
#pragma unroll
    for (int r = 0; r < 8; ++r)
      sbias[rt * 256 + (r + 8 * half) * 16 + l16] = c[r];   // [rt][j_off][h]
  }
  __syncthreads();

  // ---- stage 2: heads h = w, w+8 (waves 4..7 only do h = w) ----
  for (int h = w; h < 12; h += 8) {
    const float g = gamma[h];
    const float sp = (g > 20.f) ? g : log1pf(expf(g));
    const float coef = -0.5f * kWC * sp;

    v8f cqk = {};
    const float* qrow = q + (size_t)((b * In + i0 + l16) * Hn + h) * 16;
    const float* krow = k + (size_t)((b * In + j0 + l16) * Hn + h) * 16;
#pragma unroll
    for (int kb = 0; kb < 16; kb += 4) {
      v2f av = *reinterpret_cast<const v2f*>(qrow + kb + 2 * half);
      v2f bv = *reinterpret_cast<const v2f*>(krow + kb + 2 * half);
      cqk = WMMA_F32_4(av, bv, cqk);
    }
    v8f cpt = {};
    const float* qp = qpts + (size_t)((b * In + i0 + l16) * Hn + h) * 12;
    const float* kp = kpts + (size_t)((b * In + j0 + l16) * Hn + h) * 12;
#pragma unroll
    for (int kb = 0; kb < 12; kb += 4) {
      v2f av = *reinterpret_cast<const v2f*>(qp + kb + 2 * half);
      v2f bv = *reinterpret_cast<const v2f*>(kp + kb + 2 * half);
      cpt = WMMA_F32_4(av, bv, cpt);
    }
    const float k2v = k2[(b * In + j0 + l16) * Hn + h];
    const int arow0 = (b * Hn + h) * In + i0;
#pragma unroll
    for (int r = 0; r < 8; ++r) {
      const int m = r + 8 * half;
      const float q2v = q2[(b * In + i0 + m) * Hn + h];
      const float bias = sbias[m * 256 + l16 * 16 + h];
      const float d2 = q2v + k2v - 2.f * cpt[r];
      a[(size_t)(arow0 + m) * In + j0 + l16] = kWL * (kCSC * cqk[r] + bias + coef * d2);
    }
  }
}

// ---------------------------------------------------------------------------
// K3: column softmax over i (axis=-2). 4 threads per column, 64 columns/block.
// ---------------------------------------------------------------------------
__global__ __launch_bounds__(256) void k_softmax_col(float* __restrict__ a)
{
  __shared__ float sred[4][64];
  const int blk = blockIdx.x;            // b*H*16 + h*16 + jg
  const int jg = blk & 15;
  const int h = (blk >> 4) % 12;
  const int b = blk / (16 * 12);
  const int jcol = threadIdx.x & 63;
  const int qsec = threadIdx.x >> 6;     // i-range quarter
  const int j = jg * 64 + jcol;
  const int base = (b * Hn + h) * In + qsec * 256;

  float m = -3.4e38f;
  for (int ii = 0; ii < 256; ++ii) m = fmaxf(m, a[(size_t)(base + ii) * In + j]);
  sred[qsec][jcol] = m; __syncthreads();
  const float M = fmaxf(fmaxf(sred[0][jcol], sred[1][jcol]),
                        fmaxf(sred[2][jcol], sred[3][jcol]));
  __syncthreads();
  float ssum = 0.f;
  for (int ii = 0; ii < 256; ++ii) ssum += expf(a[(size_t)(base + ii) * In + j] - M);
  sred[qsec][jcol] = ssum; __syncthreads();
  const float inv = 1.f / (sred[0][jcol] + sred[1][jcol] + sred[2][jcol] + sred[3][jcol]);
  for (int ii = 0; ii < 256; ++ii) {
    const size_t ad = (size_t)(base + ii) * In + j;
    a[ad] = expf(a[ad] - M) * inv;
  }
}

// ---------------------------------------------------------------------------
// K4: rowsum over j of softmaxed a (o = rowsum(a)*v). One block per row.
// ---------------------------------------------------------------------------
__global__ __launch_bounds__(256) void k_rowsum(const float* __restrict__ a,
                                                float* __restrict__ rowsum)
{
  __shared__ float red[256];
  const int row = blockIdx.x;            // (b*H + h)*In + i
  const float* p = a + (size_t)row * In;
  float acc = 0.f;
  for (int u = threadIdx.x; u < In; u += 256) acc += p[u];
  red[threadIdx.x] = acc; __syncthreads();
  for (int s = 128; s > 0; s >>= 1) {
    if (threadIdx.x < s) red[threadIdx.x] += red[threadIdx.x + s];
    __syncthreads();
  }
  if (threadIdx.x == 0) rowsum[row] = red[0];
}

// ---------------------------------------------------------------------------
// K5: o_pts = a @ v_pts per (b,h). One wave per M-tile, both 16-col N-tiles
// accumulated from a single A stream (halves reads of `a`). Prefetch ahead.
// ---------------------------------------------------------------------------
__global__ __launch_bounds__(256) void k_opts(const float* __restrict__ a,
                                              const float* __restrict__ vpts,
                                              float* __restrict__ opts)
{
  const int wave = blockIdx.x * 8 + (threadIdx.x >> 5);
  const int lane = threadIdx.x & 31;
  const int half = lane >> 4, l16 = lane & 15;
  const int mt = wave & 63;
  const int h = (wave >> 6) % 12;
  const int b = wave / (64 * 12);
  const int i0 = mt * 16;

  v8f c0 = {}, c1 = {};
  const float* arow = a + (size_t)((b * Hn + h) * In + i0 + l16) * In;
  const float* vbase = vpts + (size_t)(b * In * Hn + h) * 32 + l16;
  for (int jb0 = 0; jb0 < In; jb0 += 32) {
    __builtin_prefetch(arow + jb0 + 32, 0, 0);
#pragma unroll
    for (int u = 0; u < 32; u += 4) {
      const int j = jb0 + u + 2 * half;
      v2f av = *reinterpret_cast<const v2f*>(arow + j);
      const float* vp0 = vbase + (size_t)j * (Hn * 32);
      const float* vp1 = vp0 + Hn * 32;
      v2f bv0 = { vp0[0],  vp1[0]  };
      v2f bv1 = { vp0[16], vp1[16] };
      c0 = WMMA_F32_4(av, bv0, c0);
      c1 = WMMA_F32_4(av, bv1, c1);
    }
  }
#pragma unroll
  for (int r = 0; r < 8; ++r) {
    const int m = r + 8 * half;
    const size_t obase = (size_t)((b * Hn + h) * In + i0 + m) * 32;
    opts[obase + l16]      = c0[r];
    opts[obase + 16 + l16] = c1[r];
  }
}

// ---------------------------------------------------------------------------
// K6: o_pair[b,i] = a[b,:,i,:] @ z[b,i,:,:]  (12x1024 @ 1024x16 per (b,i))
// Branch-free lane masking; 8-deep unrolled WMMA chain with prefetch.
// ---------------------------------------------------------------------------
__global__ __launch_bounds__(256) void k_opair(const float* __restrict__ a,
                                               const float* __restrict__ z,
                                               float* __restrict__ opair)
{
  const int wave = blockIdx.x * 8 + (threadIdx.x >> 5);
  const int lane = threadIdx.x & 31;
  const int half = lane >> 4, l16 = lane & 15;
  const int b = wave / In, i = wave % In;

  const int hrow = (l16 < 12) ? l16 : 11;
  const float msk = (l16 < 12) ? 1.f : 0.f;
  v8f c = {};
  const float* arow = a + (size_t)((b * Hn + hrow) * In + i) * In;
  const float* zrow = z + (size_t)(b * In + i) * In * 16;
  for (int jb0 = 0; jb0 < In; jb0 += 32) {
    __builtin_prefetch(arow + jb0 + 32, 0, 0);
#pragma unroll
    for (int u = 0; u < 32; u += 4) {
      const int j = jb0 + u + 2 * half;
      v2f av = *reinterpret_cast<const v2f*>(arow + j);
      av.x *= msk; av.y *= msk;
      v2f bv;
      bv.x = zrow[(size_t)j * 16 + l16];
      bv.y = zrow[(size_t)(j + 1) * 16 + l16];
      c = WMMA_F32_4(av, bv, c);
    }
  }
#pragma unroll
  for (int r = 0; r < 8; ++r) {
    const int hh = r + 8 * half;
    if (hh < 12) opair[(size_t)((b * In + i) * Hn + hh) * 16 + l16] = c[r];
  }
}

// ---------------------------------------------------------------------------
// K7: concat [o | o_pair | o_pts_flat | o_pts_norm] and project with Wout.
// ---------------------------------------------------------------------------
__global__ __launch_bounds__(256) void k_final(
    const float* __restrict__ v, const float* __restrict__ rowsum,
    const float* __restrict__ opair, const float* __restrict__ opts,
    const float* __restrict__ Wout, const float* __restrict__ bout,
    float* __restrict__ out)
{
  __shared__ float cat[768];
  __shared__ float red[16][16];
  const int bi = blockIdx.x;
  const int b = bi / In, i = bi % In;
  const int tid = threadIdx.x;
  for (int idx = tid; idx < 768; idx += 256) {
    float val;
    if (idx < 192) {
      const int h = idx >> 4, cc = idx & 15;
      val = rowsum[(b * Hn + h) * In + i] * v[(size_t)(bi * Hn + h) * 16 + cc];
    } else if (idx < 384) {
      val = opair[(size_t)bi * 192 + (idx - 192)];
    } else if (idx < 672) {
      const int kk = idx - 384;
      const int h = kk / 24, rest = kk % 24;
      val = opts[(size_t)((b * Hn + h) * In + i) * 32 + rest];
    } else {
      const int kk = idx - 672;
      const int h = kk >> 3, p = kk & 7;
      const float* o3 = &opts[(size_t)((b * Hn + h) * In + i) * 32 + p * 3];
      const float s2 = o3[0] * o3[0] + o3[1] * o3[1] + o3[2] * o3[2];
      val = sqrtf(fmaxf(s2, 1e-12f));
    }
    cat[idx] = val;
  }
  __syncthreads();
  const int cc = tid & 15, seg = tid >> 4;
  float acc = 0.f;
#pragma unroll 8
  for (int u = 0; u < 48; ++u) {
    const int rowi = seg * 48 + u;
    acc += cat[rowi] * Wout[rowi * 16 + cc];
  }
  red[seg][cc] = acc;
  __syncthreads();
  if (tid < 16) {
    float s = bout[tid];
#pragma unroll
    for (int g = 0; g < 16; ++g) s += red[g][tid];
    out[bi * 16 + tid] = s;
  }
}

// ---------------------------------------------------------------------------
extern "C" void kernel_launch(void* const* d_in, const int* in_sizes, int n_in,
                              void* d_out, int out_size, void* d_ws, size_t ws_size,
                              hipStream_t stream)
{
  (void)in_sizes; (void)n_in; (void)out_size; (void)ws_size;
  const float* s    = (const float*)d_in[0];
  const float* z    = (const float*)d_in[1];
  const float* R    = (const float*)d_in[2];
  const float* t    = (const float*)d_in[3];
  const float* Wq   = (const float*)d_in[4];
  const float* Wk   = (const float*)d_in[5];
  const float* Wv   = (const float*)d_in[6];
  const float* Wb   = (const float*)d_in[7];
  const float* Wqp  = (const float*)d_in[8];
  const float* Wkp  = (const float*)d_in[9];
  const float* Wvp  = (const float*)d_in[10];
  const float* gam  = (const float*)d_in[11];
  const float* Wout = (const float*)d_in[12];
  const float* bout = (const float*)d_in[13];
  float* out = (float*)d_out;

  float* ws = (float*)d_ws;
  size_t off = 0;
  float* q      = ws + off; off += (size_t)Bn * In * Hn * 16;   // 393216
  float* k      = ws + off; off += (size_t)Bn * In * Hn * 16;
  float* v      = ws + off; off += (size_t)Bn * In * Hn * 16;
  float* qpts   = ws + off; off += (size_t)Bn * In * Hn * 12;   // 294912
  float* kpts   = ws + off; off += (size_t)Bn * In * Hn * 12;
  float* vpts   = ws + off; off += (size_t)Bn * In * Hn * 32;   // 786432 (padded)
  float* q2     = ws + off; off += (size_t)Bn * In * Hn;
  float* k2     = ws + off; off += (size_t)Bn * In * Hn;
  float* rowsum = ws + off; off += (size_t)Bn * In * Hn;
  float* opts   = ws + off; off += (size_t)Bn * Hn * In * 32;   // 786432
  float* opair  = ws + off; off += (size_t)Bn * In * Hn * 16;
  float* a      = ws + off; off += (size_t)Bn * Hn * In * In;   // 25165824

  k_proj<<<Bn * In, 256, 0, stream>>>(s, R, t, Wq, Wk, Wv, Wqp, Wkp, Wvp,
                                      q, k, v, qpts, kpts, vpts, q2, k2);
  k_logits_fused<<<Bn * 64 * 64, 256, 0, stream>>>(z, Wb, q, k, qpts, kpts,
                                                   q2, k2, gam, a);
  k_softmax_col<<<Bn * Hn * 16, 256, 0, stream>>>(a);
  k_rowsum<<<Bn * Hn * In, 256, 0, stream>>>(a, rowsum);
  k_opts<<<(Bn * Hn * 64) / 8, 256, 0, stream>>>(a, vpts, opts);
  k_opair<<<(Bn * In) / 8, 256, 0, stream>>>(a, z, opair);
  k_final<<<Bn * In, 256, 0, stream>>>(v, rowsum, opair, opts, Wout, bout, out);
}